// SNMTDecoder_65025804861561
// MI455X (gfx1250) — compile-verified
//
#include <hip/hip_runtime.h>

// Problem constants (match reference).
constexpr int Bc    = 512;   // batch
constexpr int Tt    = 128;   // timesteps
constexpr int Hh    = 512;   // hidden
constexpr int Ff    = 512;   // head width
constexpr int FOURH = 4 * Hh;
constexpr int CHUNK = 4096;  // head row-chunk (scratch: 2*CHUNK*Ff*4 = 16 MB)

typedef __attribute__((ext_vector_type(2))) float v2f;
typedef __attribute__((ext_vector_type(8))) float v8f;

__device__ __forceinline__ float sigmoidf_(float x) {
  return 1.0f / (1.0f + __expf(-x));
}

#define WMMA_F32(A, Bf, C) \
  __builtin_amdgcn_wmma_f32_16x16x4_f32(false, (A), false, (Bf), (short)0, (C), false, false)

__global__ void init_counter_kernel(unsigned* cnt) {
  if (threadIdx.x == 0) *cnt = 0u;
}

// Persistent LSTM recurrence: all 128 timesteps in one kernel with a
// device-wide generation barrier between steps (grid = 128 blocks, small
// enough to be fully co-resident on any gfx1250 part).
// Block = 8 waves; wave w -> col tile; block -> 16 batch rows; each wave
// accumulates all 4 gate tiles so the cell update is wave-local.
// A tile is double-buffered in LDS: the next chunk's global load is issued
// before the 64-WMMA block and stored after it (one barrier per chunk).
__global__ __launch_bounds__(256) void lstm_persistent_kernel(
    const float* __restrict__ ctx,    // [B,T,H]
    const float* __restrict__ enc,    // [B,H]
    float* __restrict__ cbuf,         // [B,H]
    const float* __restrict__ Wx,     // [H,4H]
    const float* __restrict__ Wh,     // [H,4H]
    const float* __restrict__ bias,   // [4H]
    float* __restrict__ h0,           // [B,H] ping
    float* __restrict__ h1,           // [B,H] pong
    float* __restrict__ hglob,        // [B,T,H] decode_h in d_out
    unsigned* __restrict__ cnt)       // barrier counter (init 0)
{
  __shared__ float ldsT[2][64][17];   // K-major transposed A tile, padded

  const int lane = threadIdx.x & 31;
  const int m16  = lane & 15;               // M (A) / N (B) within tile
  const int koff = (lane >> 4) << 1;        // lanes 16-31 hold K+2
  const int hi8  = (lane >> 4) << 3;        // C/D row offset for upper lanes
  const int wave = threadIdx.x >> 5;
  const int rowBase = (blockIdx.x >> 2) << 4;              // batch row tile
  const int c0 = (((blockIdx.x & 3) << 3) + wave) << 4;    // hidden-col tile

  const int mm = threadIdx.x >> 4;          // staging: row 0..15
  const int ks = (threadIdx.x & 15) << 2;   // staging: k 0..60 step 4

  const int col = c0 + m16;
  const float bi = bias[0 * Hh + col];
  const float bf = bias[1 * Hh + col];
  const float bg = bias[2 * Hh + col];
  const float bo = bias[3 * Hh + col];

  for (int t = 0; t < Tt; ++t) {
    const float* __restrict__ hprev = (t == 0) ? enc : ((t & 1) ? h0 : h1);
    float* __restrict__ hcur = (t & 1) ? h1 : h0;

    v8f acc0 = {}, acc1 = {}, acc2 = {}, acc3 = {};

    // Staging source for chunk c (c = 0..15: phase = c>>3, k0 = (c&7)*64).
    const float* srcX = ctx + ((size_t)(rowBase + mm) * Tt + t) * Hh + ks;
    const float* srcH = hprev + (size_t)(rowBase + mm) * Hh + ks;

    // Prologue: stage chunk 0.
    {
      float4 v = *(const float4*)srcX;
      ldsT[0][ks + 0][mm] = v.x;
      ldsT[0][ks + 1][mm] = v.y;
      ldsT[0][ks + 2][mm] = v.z;
      ldsT[0][ks + 3][mm] = v.w;
    }
    __syncthreads();

    for (int c = 0; c < 16; ++c) {
      const int phase = c >> 3;
      const int k0 = (c & 7) << 6;
      const float* __restrict__ Wp = phase ? Wh : Wx;
      const int buf = c & 1;

      // Issue next chunk's global load early (lands in LDS after compute).
      float4 vnext;
      if (c + 1 < 16) {
        const int np = (c + 1) >> 3;
        const int nk = ((c + 1) & 7) << 6;
        vnext = *(const float4*)((np ? srcH : srcX) + nk);
      }
      __builtin_prefetch(Wp + (size_t)(k0 + 64) * FOURH + (c0 + m16), 0, 3);

      for (int kk = 0; kk < 64; kk += 4) {
        v2f a;
        a.x = ldsT[buf][kk + koff + 0][m16];
        a.y = ldsT[buf][kk + koff + 1][m16];
        const float* wrow = Wp + (size_t)(k0 + kk + koff) * FOURH + (c0 + m16);
        v2f bi_, bf_, bg_, bo_;
        bi_.x = wrow[0 * Hh]; bi_.y = wrow[0 * Hh + FOURH];
        bf_.x = wrow[1 * Hh]; bf_.y = wrow[1 * Hh + FOURH];
        bg_.x = wrow[2 * Hh]; bg_.y = wrow[2 * Hh + FOURH];
        bo_.x = wrow[3 * Hh]; bo_.y = wrow[3 * Hh + FOURH];
        acc0 = WMMA_F32(a, bi_, acc0);
        acc1 = WMMA_F32(a, bf_, acc1);
        acc2 = WMMA_F32(a, bg_, acc2);
        acc3 = WMMA_F32(a, bo_, acc3);
      }

      if (c + 1 < 16) {
        ldsT[buf ^ 1][ks + 0][mm] = vnext.x;
        ldsT[buf ^ 1][ks + 1][mm] = vnext.y;
        ldsT[buf ^ 1][ks + 2][mm] = vnext.z;
        ldsT[buf ^ 1][ks + 3][mm] = vnext.w;
      }
      __syncthreads();
    }

    // Fused LSTM cell update on this wave's 16x16 tile.
#pragma unroll
    for (int i = 0; i < 8; ++i) {
      const int rb = rowBase + hi8 + i;
      const float zi = acc0[i] + bi;
      const float zf = acc1[i] + bf;
      const float zg = acc2[i] + bg;
      const float zo = acc3[i] + bo;
      const size_t cidx = (size_t)rb * Hh + col;
      const float cold = (t == 0) ? 0.0f : cbuf[cidx];
      const float cn = sigmoidf_(zf) * cold + sigmoidf_(zi) * tanhf(zg);
      const float hn = sigmoidf_(zo) * tanhf(cn);
      cbuf[cidx] = cn;
      hcur[cidx] = hn;
      hglob[((size_t)rb * Tt + t) * Hh + col] = hn;
    }

    // Device-wide barrier: generation counter, no reset race.
    if (t != Tt - 1) {
      __threadfence();
      __syncthreads();
      if (threadIdx.x == 0) {
        __hip_atomic_fetch_add(cnt, 1u, __ATOMIC_RELEASE, __HIP_MEMORY_SCOPE_AGENT);
        const unsigned target = (unsigned)(t + 1) * (unsigned)gridDim.x;
        while (__hip_atomic_load(cnt, __ATOMIC_ACQUIRE, __HIP_MEMORY_SCOPE_AGENT) < target)
          __builtin_amdgcn_s_sleep(2);
      }
      __syncthreads();
      __threadfence();
    }
  }
}

// out[r, n] = relu(A[r, :K] @ W[:, n] + bias[n]).
// Each wave owns one 16-col tile and TWO 16-row tiles (B fragments reused
// across both -> halves L2 weight traffic). Pad 41 keeps LDS stores and the
// dual A-fragment reads bank-conflict-free. A tile double-buffered.
__global__ __launch_bounds__(256) void dense_relu_kernel(
    const float* __restrict__ A, const float* __restrict__ W,
    const float* __restrict__ bias, float* __restrict__ out, int K, int N)
{
  __shared__ float ldsT[2][64][41];
  const int lane = threadIdx.x & 31;
  const int m16  = lane & 15;
  const int koff = (lane >> 4) << 1;
  const int hi8  = (lane >> 4) << 3;
  const int wave = threadIdx.x >> 5;
  const int rowBase = blockIdx.y << 5;                 // 32 rows per block
  const int c0 = ((blockIdx.x << 3) + wave) << 4;

  v8f acc0 = {}, acc1 = {};
  const int mm = threadIdx.x >> 3;          // staging: row 0..31
  const int ks = (threadIdx.x & 7) << 3;    // staging: k 0..56 step 8

  const float* srcA = A + (size_t)(rowBase + mm) * K + ks;
  const int nchunk = K >> 6;

  // Prologue: stage chunk 0.
  {
    float4 v0 = *(const float4*)srcA;
    float4 v1 = *(const float4*)(srcA + 4);
    ldsT[0][ks + 0][mm] = v0.x; ldsT[0][ks + 1][mm] = v0.y;
    ldsT[0][ks + 2][mm] = v0.z; ldsT[0][ks + 3][mm] = v0.w;
    ldsT[0][ks + 4][mm] = v1.x; ldsT[0][ks + 5][mm] = v1.y;
    ldsT[0][ks + 6][mm] = v1.z; ldsT[0][ks + 7][mm] = v1.w;
  }
  __syncthreads();

  for (int c = 0; c < nchunk; ++c) {
    const int k0 = c << 6;
    const int buf = c & 1;

    float4 v0n, v1n;
    if (c + 1 < nchunk) {
      v0n = *(const float4*)(srcA + (k0 + 64));
      v1n = *(const float4*)(srcA + (k0 + 68));
    }
    __builtin_prefetch(W + (size_t)(k0 + 64) * N + (c0 + m16), 0, 3);

    for (int kk = 0; kk < 64; kk += 4) {
      v2f a0, a1;
      a0.x = ldsT[buf][kk + koff + 0][m16];
      a0.y = ldsT[buf][kk + koff + 1][m16];
      a1.x = ldsT[buf][kk + koff + 0][16 + m16];
      a1.y = ldsT[buf][kk + koff + 1][16 + m16];
      const float* wrow = W + (size_t)(k0 + kk + koff) * N + (c0 + m16);
      v2f bfr;
      bfr.x = wrow[0];
      bfr.y = wrow[N];
      acc0 = WMMA_F32(a0, bfr, acc0);
      acc1 = WMMA_F32(a1, bfr, acc1);
    }

    if (c + 1 < nchunk) {
      ldsT[buf ^ 1][ks + 0][mm] = v0n.x; ldsT[buf ^ 1][ks + 1][mm] = v0n.y;
      ldsT[buf ^ 1][ks + 2][mm] = v0n.z; ldsT[buf ^ 1][ks + 3][mm] = v0n.w;
      ldsT[buf ^ 1][ks + 4][mm] = v1n.x; ldsT[buf ^ 1][ks + 5][mm] = v1n.y;
      ldsT[buf ^ 1][ks + 6][mm] = v1n.z; ldsT[buf ^ 1][ks + 7][mm] = v1n.w;
    }
    __syncthreads();
  }

  const int col = c0 + m16;
  const float bb = bias[col];
#pragma unroll
  for (int i = 0; i < 8; ++i) {
    out[(size_t)(rowBase + hi8 + i) * N + col]      = fmaxf(acc0[i] + bb, 0.0f);
    out[(size_t)(rowBase + 16 + hi8 + i) * N + col] = fmaxf(acc1[i] + bb, 0.0f);
  }
}

// y[r] = X[r,:] . W3 + b3  (one wave per row, float4 coalesced loads,
// shfl reduction -> deterministic)
__global__ __launch_bounds__(128) void head_dot_kernel(
    const float* __restrict__ X, const float* __restrict__ W3,
    const float* __restrict__ b3, float* __restrict__ y)
{
  const int r = (blockIdx.x << 2) + (threadIdx.x >> 5);
  const int lane = threadIdx.x & 31;
  float s = 0.0f;
#pragma unroll
  for (int j0 = 0; j0 < Ff; j0 += 128) {
    float4 xv = *(const float4*)(X + (size_t)r * Ff + j0 + (lane << 2));
    float4 wv = *(const float4*)(W3 + j0 + (lane << 2));
    s += xv.x * wv.x + xv.y * wv.y + xv.z * wv.z + xv.w * wv.w;
  }
#pragma unroll
  for (int m = 16; m; m >>= 1) s += __shfl_xor(s, m, 32);
  if (lane == 0) y[r] = s + b3[0];
}

extern "C" void kernel_launch(void* const* d_in, const int* in_sizes, int n_in,
                              void* d_out, int out_size, void* d_ws, size_t ws_size,
                              hipStream_t stream)
{
  const float* ctx = (const float*)d_in[0];   // [B,T,H]
  const float* enc = (const float*)d_in[1];   // [B,H]
  const float* Wx  = (const float*)d_in[2];   // [H,4H]
  const float* Wh  = (const float*)d_in[3];   // [H,4H]
  const float* b   = (const float*)d_in[4];   // [4H]
  const float* W1  = (const float*)d_in[5];   // [H,F]
  const float* b1  = (const float*)d_in[6];   // [F]
  const float* W2  = (const float*)d_in[7];   // [F,F]
  const float* b2  = (const float*)d_in[8];   // [F]
  const float* W3  = (const float*)d_in[9];   // [F,1]
  const float* b3  = (const float*)d_in[10];  // [1]
  (void)in_sizes; (void)n_in; (void)out_size; (void)ws_size;

  float* traj  = (float*)d_out;               // [B,T,1] flat
  float* hglob = traj + (size_t)Bc * Tt;      // [B,T,H] flat

  float* ws    = (float*)d_ws;
  float* cbuf  = ws;                          // B*H
  float* h0    = cbuf + (size_t)Bc * Hh;      // B*H ping
  float* h1    = h0 + (size_t)Bc * Hh;        // B*H pong
  float* x1buf = h1 + (size_t)Bc * Hh;        // CHUNK*F
  float* x2buf = x1buf + (size_t)CHUNK * Ff;  // CHUNK*F
  unsigned* cnt = (unsigned*)(x2buf + (size_t)CHUNK * Ff);

  // Recurrence: one persistent kernel, device-wide barrier between steps.
  init_counter_kernel<<<1, 64, 0, stream>>>(cnt);
  lstm_persistent_kernel<<<128, 256, 0, stream>>>(
      ctx, enc, cbuf, Wx, Wh, b, h0, h1, hglob, cnt);

  // Batched 3-layer head over all B*T rows (chunked for scratch).
  for (int r0 = 0; r0 < Bc * Tt; r0 += CHUNK) {
    dense_relu_kernel<<<dim3(4, CHUNK / 32), 256, 0, stream>>>(
        hglob + (size_t)r0 * Hh, W1, b1, x1buf, Hh, Ff);
    dense_relu_kernel<<<dim3(4, CHUNK / 32), 256, 0, stream>>>(
        x1buf, W2, b2, x2buf, Ff, Ff);
    head_dot_kernel<<<CHUNK / 4, 128, 0, stream>>>(x2buf, W3, b3, traj + r0);
  }
}